// SharedRadialLinearTransform_60567628808530
// MI455X (gfx1250) — compile-verified
//
#include <hip/hip_runtime.h>

typedef __attribute__((ext_vector_type(2))) float v2f;
typedef __attribute__((ext_vector_type(8))) float v8f;
typedef __attribute__((ext_vector_type(4))) unsigned int v4u;
typedef __attribute__((ext_vector_type(8))) int v8i;
typedef __attribute__((ext_vector_type(4))) int v4i;

#define N_NODES 4000
#define RDIM 256
#define MDIM 256
#define ADIM 20
#define EDIM 8
#define KSTEPS (RDIM / 4)       // 64 k-steps of K=4
#define MTILES (MDIM / 16)      // 16 m-tiles of 16
#define NBLK 25                 // node blocks (grid.x)
#define ITERS 10                // 16-node tiles per block: 25*10*16 = 4000
#define LDS_STRIDE 264          // 256 + 8 pad floats per row (TDM pad) -> conflict-free
#define LDS_FLOATS (RDIM * LDS_STRIDE)   // 270,336 B of LDS (<= 320 KB)

// out[n,m,a,e] = sum_r x[n,r,a,e] * W_{g(a)}[r,m]
__global__ void __launch_bounds__(256, 1)
srlt_wmma_kernel(const float* __restrict__ x,
                 const float* __restrict__ w,
                 float* __restrict__ out) {
    extern __shared__ float smem[];

    const int a = blockIdx.y;                                   // angular index 0..19
    const int g = (a >= 10) ? 3 : ((a >= 4) ? 2 : ((a >= 1) ? 1 : 0));
    const float* __restrict__ W = w + (size_t)g * RDIM * MDIM;  // [r][m], row-major

    // ---- Stage W_g (256x256 f32) into LDS via the Tensor Data Mover ----
    // D# group0: count=1, lds_addr, global_addr, type=2 ("image")
    // D# group1: data_size=4B, pad_enable, pad_interval=256 DW (row), pad_amount=8 DW,
    //            tensor_dim0=256, tensor_dim1=256, tile_dim0=256, tile_dim1=256,
    //            tensor_dim0_stride=256 elements.
    // Result: LDS image lds_w[r*264 + m], exactly our padded layout.
    if ((threadIdx.x >> 5) == 0) {   // wave 0 issues the DMA
        const unsigned long long gaddr = (unsigned long long)(uintptr_t)W;
        const unsigned int ldsoff = (unsigned int)(uintptr_t)smem;  // LDS byte offset
        v4u g0;
        g0[0] = 1u;                                            // count=1, user mode
        g0[1] = ldsoff;                                        // lds_addr (bytes)
        g0[2] = (unsigned int)(gaddr & 0xFFFFFFFFu);           // global_addr[31:0]
        g0[3] = (unsigned int)((gaddr >> 32) & 0x01FFFFFFu)    // global_addr[56:32]
              | (2u << 30);                                    // type=2
        v8i g1;
        g1[0] = (int)((2u << 16)      // data_size = 4 bytes
                    | (1u << 20)      // pad_enable
                    | (7u << 22)      // pad_interval: 256 DWORDs (one row)
                    | (7u << 25));    // pad_amount: 8 DWORDs
        g1[1] = (int)(256u << 16);    // tensor_dim0[15:0] in bits 63:48
        g1[2] = (int)(256u << 16);    // tensor_dim1[15:0] in bits 95:80
        g1[3] = (int)(256u << 16);    // tile_dim0 = 256 (bits 127:112)
        g1[4] = 256;                  // tile_dim1 = 256 (bits 143:128), tile_dim2 = 0
        g1[5] = 256;                  // tensor_dim0_stride[31:0] = 256 elements
        g1[6] = 0;                    // stride[47:32]=0, dim1_stride[15:0]=0
        g1[7] = 1;                    // dim1_stride[47:16] -> 65536 (unused, tile_dim2=0)
        v4i gz = {0, 0, 0, 0};
#if __clang_major__ >= 23
        v8i gz8 = {0, 0, 0, 0, 0, 0, 0, 0};
        __builtin_amdgcn_tensor_load_to_lds(g0, g1, gz, gz, gz8, 0);
#else
        __builtin_amdgcn_tensor_load_to_lds(g0, g1, gz, gz, 0);
#endif
        __builtin_amdgcn_s_wait_tensorcnt(0);
    }
    __syncthreads();

    const int lane = threadIdx.x & 31;
    const int wv   = threadIdx.x >> 5;        // 8 waves: distinct node pairs

    // A frag (16x4 f32): lanes0-15 row M=lane (K=0/1), lanes16-31 row M=lane-16 (K=2/3)
    const int Mrow  = lane & 15;              // row = node_local*8 + e
    const int eA    = Mrow & 7;
    const int nAoff = Mrow >> 3;              // 0/1 inside node pair
    const int rOff  = (lane >> 4) << 1;       // 0 or 2 (K half)

    // B frag (4x16 f32): VGPR0 lanes0-15 = row K=0 (+0), lanes16-31 = row K=2 (+2*264);
    //                    VGPR1 = +1 row (+264). Column = 16*mt + (lane&15).
    const float* __restrict__ bl = smem + (size_t)((lane >> 4) * 2) * LDS_STRIDE + (lane & 15);

    for (int it = 0; it < ITERS; ++it) {
        const int n0 = (blockIdx.x * ITERS + it) * 16 + wv * 2;   // this wave's node pair

        const float* __restrict__ xb =
            x + (((size_t)(n0 + nAoff) * RDIM + rOff) * ADIM + a) * EDIM + eA;

        v8f acc[MTILES];
        #pragma unroll
        for (int mt = 0; mt < MTILES; ++mt)
            acc[mt] = (v8f){0.f, 0.f, 0.f, 0.f, 0.f, 0.f, 0.f, 0.f};

        for (int k = 0; k < KSTEPS; ++k) {
            const float a0 = xb[(size_t)(4 * k + 0) * (ADIM * EDIM)];
            const float a1 = xb[(size_t)(4 * k + 1) * (ADIM * EDIM)];
            const v2f Af = {a0, a1};

            const float* __restrict__ bk = bl + (size_t)k * 4 * LDS_STRIDE;
            #pragma unroll
            for (int mt = 0; mt < MTILES; ++mt) {
                const v2f Bf = {bk[mt * 16], bk[mt * 16 + LDS_STRIDE]};
                acc[mt] = __builtin_amdgcn_wmma_f32_16x16x4_f32(
                    /*neg_a=*/false, Af, /*neg_b=*/false, Bf,
                    /*c_mod=*/(short)0, acc[mt],
                    /*reuse_a=*/false, /*reuse_b=*/false);
            }
        }

        // C/D: VGPR v, lanes<16 -> M=v (node0,e=v); lanes>=16 -> M=v+8 (node1,e=v).
        // Each lane owns out[n0+(lane>>4)][m0+(lane&15)][a][0..7]: 32B contiguous.
        float* __restrict__ ob =
            out + (((size_t)(n0 + (lane >> 4)) * MDIM + (lane & 15)) * ADIM + a) * EDIM;
        #pragma unroll
        for (int mt = 0; mt < MTILES; ++mt) {
            float4 lo, hi;
            lo.x = acc[mt][0]; lo.y = acc[mt][1]; lo.z = acc[mt][2]; lo.w = acc[mt][3];
            hi.x = acc[mt][4]; hi.y = acc[mt][5]; hi.z = acc[mt][6]; hi.w = acc[mt][7];
            float* p = ob + (size_t)mt * 16 * (ADIM * EDIM);  // m-tile step: 16 m * 160 floats
            *(float4*)(p + 0) = lo;
            *(float4*)(p + 4) = hi;
        }
    }
}

extern "C" void kernel_launch(void* const* d_in, const int* in_sizes, int n_in,
                              void* d_out, int out_size, void* d_ws, size_t ws_size,
                              hipStream_t stream) {
    (void)in_sizes; (void)n_in; (void)out_size; (void)d_ws; (void)ws_size;
    const float* x = (const float*)d_in[0];   // [4000, 256, 20, 8]
    const float* w = (const float*)d_in[1];   // [4, 256, 256]
    float* out = (float*)d_out;               // [4000, 256, 20, 8]

    dim3 grid(NBLK, ADIM, 1);   // 25 node-blocks x 20 angular indices
    dim3 block(256, 1, 1);      // 8 waves
    size_t lds_bytes = (size_t)LDS_FLOATS * sizeof(float);  // 264-float padded rows
    srlt_wmma_kernel<<<grid, block, lds_bytes, stream>>>(x, w, out);
}